// GINet_10110353015120
// MI455X (gfx1250) — compile-verified
//
#include <hip/hip_runtime.h>
#include <hip/hip_bf16.h>
#include <math.h>

// ---------------------------------------------------------------------------
// GIN forward for MI455X (gfx1250, wave32).
//   EMB=300 (pad 304), 2*EMB=600 (pad 608), FEAT=512, 5 layers, 64 graphs.
//   GEMMs use V_WMMA_F32_16X16X4_F32 (full fp32), register-blocked 4 M-tiles
//   per wave so each B fragment feeds 4 WMMAs (VMEM-issue relief from L2).
// ---------------------------------------------------------------------------

#define EMB   300
#define HP    304      // padded EMB   (19 tiles of 16)
#define H2    600
#define H2P   608      // padded 2*EMB (38 tiles of 16)
#define FEAT  512
#define NG    64
#define NL    5

typedef __attribute__((ext_vector_type(2))) float v2f;
typedef __attribute__((ext_vector_type(8))) float v8f;

// ---------- small helpers ---------------------------------------------------

__global__ void zero_k(float* __restrict__ p, long n) {
  long i = blockIdx.x * (long)blockDim.x + threadIdx.x;
  if (i < n) p[i] = 0.f;
}

// dst[rows][dcols] = zero-padded src[rows][scols]
__global__ void pad_rows(const float* __restrict__ src, float* __restrict__ dst,
                         int rows, int scols, int dcols) {
  long i = blockIdx.x * (long)blockDim.x + threadIdx.x;
  long n = (long)rows * dcols;
  if (i >= n) return;
  int r = (int)(i / dcols), c = (int)(i % dcols);
  dst[i] = (c < scols) ? src[(long)r * scols + c] : 0.f;
}

// h[n][0:300] = x_emb1[x[n,0]] + x_emb2[x[n,1]] ; pad cols zeroed
__global__ void node_init(const int* __restrict__ x,
                          const float* __restrict__ e1,
                          const float* __restrict__ e2,
                          float* __restrict__ h, int N) {
  long i = blockIdx.x * (long)blockDim.x + threadIdx.x;
  if (i >= (long)N * HP) return;
  int r = (int)(i / HP), c = (int)(i % HP);
  float v = 0.f;
  if (c < EMB) v = e1[(long)x[2 * r] * EMB + c] + e2[(long)x[2 * r + 1] * EMB + c];
  h[i] = v;
}

// ---------- edge scatter (message + segment_sum) ----------------------------
// One 96-thread block per edge (E real edges + N self loops with attr (5,0)).
// float4 message construction (rows are 16B aligned), f32 hardware atomics.
__global__ void scatter_edges(const float* __restrict__ h,
                              const int* __restrict__ ei,   // [2][E]
                              const int* __restrict__ ea,   // [E][2]
                              const float* __restrict__ e1, // [6][300]
                              const float* __restrict__ e2, // [3][300]
                              float* __restrict__ agg, int N, int E) {
  int e = blockIdx.x;
  int src, dst, a0, a1;
  if (e < E) {
    src = ei[e]; dst = ei[E + e];
    a0 = ea[2 * e]; a1 = ea[2 * e + 1];
  } else {
    src = dst = e - E; a0 = 5; a1 = 0;   // self loop
  }
  int t = threadIdx.x;
  if (t < EMB / 4) {                     // 75 float4 per row
    const float4* hs = (const float4*)(h + (long)src * HP);
    const float4* p0 = (const float4*)(e1 + (long)a0 * EMB);
    const float4* p1 = (const float4*)(e2 + (long)a1 * EMB);
    float4 hv = hs[t], v0 = p0[t], v1 = p1[t];
    float* ad = agg + (long)dst * HP + 4 * t;
    unsafeAtomicAdd(&ad[0], hv.x + v0.x + v1.x);
    unsafeAtomicAdd(&ad[1], hv.y + v0.y + v1.y);
    unsafeAtomicAdd(&ad[2], hv.z + v0.z + v1.z);
    unsafeAtomicAdd(&ad[3], hv.w + v0.w + v1.w);
  }
}

// ---------- fp32 WMMA GEMM: 64x16 strip per wave ----------------------------
// C = act(A @ B + bias).  A: [M, lda] (K <= lda, M % 64 == 0),
// B: [K, ldb] zero-padded so every column load is valid.
// Each wave: 4 M-tiles x 1 N-tile; B fragment reused across 4 WMMAs.
// grid = (M/64, ceil(Ntiles/4)), block = 128 (4 waves).
// act: 0 = none, 1 = relu, 2 = softplus
__global__ __launch_bounds__(128)
void gemm_wmma_f32(const float* __restrict__ A, int lda,
                   const float* __restrict__ B, int ldb,
                   const float* __restrict__ bias,
                   float* __restrict__ C, int ldc,
                   int K, int Ntiles, int act) {
  const int wave = threadIdx.x >> 5;
  const int ntile = blockIdx.y * 4 + wave;
  if (ntile >= Ntiles) return;              // wave-uniform: EXEC stays full
  const int lane = threadIdx.x & 31;
  const int half = lane >> 4;               // K-pair select (fp32 A layout)
  const int lm   = lane & 15;
  const int row0 = blockIdx.x * 64;         // 4 M-tiles per wave
  const int col  = ntile * 16 + lm;

  const float* Ap = A + (long)(row0 + lm) * lda + 2 * half;
  const float* Bp = B + (long)(2 * half) * ldb + col;
  const long astep = 16L * lda;

  v8f c0 = {0.f,0.f,0.f,0.f,0.f,0.f,0.f,0.f};
  v8f c1 = c0, c2 = c0, c3 = c0;

  for (int k = 0; k < K; k += 4) {
    v2f b;
    b[0] = Bp[(long)k * ldb];
    b[1] = Bp[(long)(k + 1) * ldb];
    v2f a0, a1, a2, a3;
    a0[0] = Ap[k];             a0[1] = Ap[k + 1];
    a1[0] = Ap[astep + k];     a1[1] = Ap[astep + k + 1];
    a2[0] = Ap[2 * astep + k]; a2[1] = Ap[2 * astep + k + 1];
    a3[0] = Ap[3 * astep + k]; a3[1] = Ap[3 * astep + k + 1];
    c0 = __builtin_amdgcn_wmma_f32_16x16x4_f32(false, a0, false, b, (short)0, c0, false, false);
    c1 = __builtin_amdgcn_wmma_f32_16x16x4_f32(false, a1, false, b, (short)0, c1, false, false);
    c2 = __builtin_amdgcn_wmma_f32_16x16x4_f32(false, a2, false, b, (short)0, c2, false, false);
    c3 = __builtin_amdgcn_wmma_f32_16x16x4_f32(false, a3, false, b, (short)0, c3, false, false);
  }

  const float bv = bias[col];
  v8f acc[4] = {c0, c1, c2, c3};
#pragma unroll
  for (int mt = 0; mt < 4; ++mt) {
#pragma unroll
    for (int v = 0; v < 8; ++v) {
      int r = row0 + mt * 16 + v + 8 * half;   // fp32 C/D layout
      float x = acc[mt][v] + bv;
      if (act == 1) x = fmaxf(x, 0.f);
      else if (act == 2) x = (x > 20.f) ? x : log1pf(__expf(x));
      C[(long)r * ldc + col] = x;
    }
  }
}

// ---------- batch norm ------------------------------------------------------
__global__ void bn_stats(const float* __restrict__ hn, float* __restrict__ mean,
                         float* __restrict__ rstd, int N) {
  int c = blockIdx.x;                        // one column per block
  float s = 0.f, ss = 0.f;
  for (int r = threadIdx.x; r < N; r += blockDim.x) {
    float v = hn[(long)r * HP + c];
    s += v; ss += v * v;
  }
  __shared__ float sh[256], sh2[256];
  sh[threadIdx.x] = s; sh2[threadIdx.x] = ss;
  __syncthreads();
  for (int o = blockDim.x >> 1; o > 0; o >>= 1) {
    if ((int)threadIdx.x < o) {
      sh[threadIdx.x] += sh[threadIdx.x + o];
      sh2[threadIdx.x] += sh2[threadIdx.x + o];
    }
    __syncthreads();
  }
  if (threadIdx.x == 0) {
    float m = sh[0] / (float)N;
    float v = sh2[0] / (float)N - m * m;
    mean[c] = m;
    rstd[c] = rsqrtf(v + 1e-5f);
  }
}

__global__ void bn_apply(const float* __restrict__ hn,
                         const float* __restrict__ mean,
                         const float* __restrict__ rstd,
                         const float* __restrict__ g, const float* __restrict__ b,
                         float* __restrict__ h, int N, int dorelu) {
  long i = blockIdx.x * (long)blockDim.x + threadIdx.x;
  if (i >= (long)N * EMB) return;
  int r = (int)(i / EMB), c = (int)(i % EMB);
  float v = (hn[(long)r * HP + c] - mean[c]) * rstd[c] * g[c] + b[c];
  if (dorelu) v = fmaxf(v, 0.f);
  h[(long)r * HP + c] = v;
}

// ---------- global mean pool ------------------------------------------------
__global__ void pool_sum(const float* __restrict__ h, const int* __restrict__ batch,
                         float* __restrict__ sums, int N) {
  long i = blockIdx.x * (long)blockDim.x + threadIdx.x;
  if (i >= (long)N * EMB) return;
  int r = (int)(i / EMB), c = (int)(i % EMB);
  unsafeAtomicAdd(&sums[(long)batch[r] * EMB + c], h[(long)r * HP + c]);
}

__global__ void pool_cnt(const int* __restrict__ batch, float* __restrict__ cnts, int N) {
  int i = blockIdx.x * blockDim.x + threadIdx.x;
  if (i < N) unsafeAtomicAdd(&cnts[batch[i]], 1.f);
}

__global__ void pool_div(const float* __restrict__ sums, const float* __restrict__ cnts,
                         float* __restrict__ hg) {
  int i = blockIdx.x * blockDim.x + threadIdx.x;
  if (i >= NG * EMB) return;
  hg[i] = sums[i] / fmaxf(cnts[i / EMB], 1.f);
}

// ---------- final 64x2 head -------------------------------------------------
__global__ void pred_head(const float* __restrict__ p2, const float* __restrict__ w,
                          const float* __restrict__ b, float* __restrict__ out) {
  int idx = threadIdx.x;
  if (idx >= NG * 2) return;
  int g = idx >> 1, o = idx & 1;
  float s = b[o];
  for (int k = 0; k < FEAT / 2; ++k) s += p2[g * (FEAT / 2) + k] * w[k * 2 + o];
  out[g * 2 + o] = s;
}

// ---------------------------------------------------------------------------

extern "C" void kernel_launch(void* const* d_in, const int* in_sizes, int n_in,
                              void* d_out, int out_size, void* d_ws, size_t ws_size,
                              hipStream_t stream) {
  (void)n_in; (void)out_size; (void)ws_size;
  const int N = in_sizes[0] / 2;     // 16000
  const int E = in_sizes[1] / 2;     // 256000

  const int*   x      = (const int*)d_in[0];
  const int*   ei     = (const int*)d_in[1];
  const int*   ea     = (const int*)d_in[2];
  const int*   batch  = (const int*)d_in[3];
  const float* x_emb1 = (const float*)d_in[4];
  const float* x_emb2 = (const float*)d_in[5];
  const float* e1_emb = (const float*)d_in[6];
  const float* e2_emb = (const float*)d_in[7];
  const float* w1     = (const float*)d_in[8];
  const float* b1     = (const float*)d_in[9];
  const float* w2     = (const float*)d_in[10];
  const float* b2     = (const float*)d_in[11];
  const float* bn_g   = (const float*)d_in[12];
  const float* bn_b   = (const float*)d_in[13];
  const float* feat_w = (const float*)d_in[14];
  const float* feat_b = (const float*)d_in[15];
  const float* hw1    = (const float*)d_in[16];
  const float* hb1    = (const float*)d_in[17];
  const float* hw2    = (const float*)d_in[18];
  const float* hb2    = (const float*)d_in[19];
  const float* hw3    = (const float*)d_in[20];
  const float* hb3    = (const float*)d_in[21];

  float* out  = (float*)d_out;
  float* hf   = out;                 // [64][512]
  float* pred = out + NG * FEAT;     // [64][2]

  // workspace layout (floats)
  float* ws   = (float*)d_ws;
  float* h    = ws;  ws += (long)N * HP;
  float* agg  = ws;  ws += (long)N * HP;      // reused as hnew after GEMM2
  float* tmp  = ws;  ws += (long)N * H2P;
  float* w1p  = ws;  ws += (long)NL * EMB * H2P;
  float* w2p  = ws;  ws += (long)NL * H2 * HP;
  float* b1p  = ws;  ws += (long)NL * H2P;
  float* b2p  = ws;  ws += (long)NL * HP;
  float* mean = ws;  ws += HP;
  float* rstd = ws;  ws += HP;
  float* sums = ws;  ws += NG * EMB;          // cnts must follow contiguously
  float* cnts = ws;  ws += NG;
  float* hg   = ws;  ws += NG * EMB;
  float* p1   = ws;  ws += NG * (FEAT / 2);
  float* p2   = ws;  ws += NG * (FEAT / 2);

  const int T = 256;
  #define NB(n) ((int)(((long)(n) + T - 1) / T))

  // pack zero-padded weights/biases (N dims -> multiples of 16)
  pad_rows<<<NB((long)NL * EMB * H2P), T, 0, stream>>>(w1, w1p, NL * EMB, H2, H2P);
  pad_rows<<<NB((long)NL * H2 * HP),  T, 0, stream>>>(w2, w2p, NL * H2, EMB, HP);
  pad_rows<<<NB(NL * H2P), T, 0, stream>>>(b1, b1p, NL, H2, H2P);
  pad_rows<<<NB(NL * HP),  T, 0, stream>>>(b2, b2p, NL, EMB, HP);

  node_init<<<NB((long)N * HP), T, 0, stream>>>(x, x_emb1, x_emb2, h, N);

  const int Mg = N / 64;                     // 250 M-groups (64 rows each)
  for (int l = 0; l < NL; ++l) {
    zero_k<<<NB((long)N * HP), T, 0, stream>>>(agg, (long)N * HP);
    scatter_edges<<<E + N, 96, 0, stream>>>(h, ei, ea,
                                            e1_emb + (long)l * 6 * EMB,
                                            e2_emb + (long)l * 3 * EMB,
                                            agg, N, E);
    // tmp = relu(agg @ w1 + b1)   [N,300]x[300,608] -> [N,608] (38 tiles)
    gemm_wmma_f32<<<dim3(Mg, 10), 128, 0, stream>>>(
        agg, HP, w1p + (long)l * EMB * H2P, H2P, b1p + (long)l * H2P,
        tmp, H2P, EMB, H2P / 16, 1);
    // hnew = tmp @ w2 + b2        [N,600]x[600,304] -> [N,304] (19 tiles)
    gemm_wmma_f32<<<dim3(Mg, 5), 128, 0, stream>>>(
        tmp, H2P, w2p + (long)l * H2 * HP, HP, b2p + (long)l * HP,
        agg, HP, H2, HP / 16, 0);
    bn_stats<<<EMB, 256, 0, stream>>>(agg, mean, rstd, N);
    bn_apply<<<NB((long)N * EMB), T, 0, stream>>>(agg, mean, rstd,
                                                  bn_g + (long)l * EMB,
                                                  bn_b + (long)l * EMB,
                                                  h, N, l < NL - 1);
  }

  // global mean pool
  zero_k<<<NB(NG * EMB + NG), T, 0, stream>>>(sums, NG * EMB + NG);
  pool_sum<<<NB((long)N * EMB), T, 0, stream>>>(h, batch, sums, N);
  pool_cnt<<<NB(N), T, 0, stream>>>(batch, cnts, N);
  pool_div<<<NB(NG * EMB), T, 0, stream>>>(sums, cnts, hg);

  // hf = hg @ feat_w + feat_b   [64,300]x[300,512]
  gemm_wmma_f32<<<dim3(1, 8), 128, 0, stream>>>(
      hg, EMB, feat_w, FEAT, feat_b, hf, FEAT, EMB, FEAT / 16, 0);
  // p1 = softplus(hf @ hw1 + hb1)   [64,512]x[512,256]
  gemm_wmma_f32<<<dim3(1, 4), 128, 0, stream>>>(
      hf, FEAT, hw1, FEAT / 2, hb1, p1, FEAT / 2, FEAT, FEAT / 32, 2);
  // p2 = softplus(p1 @ hw2 + hb2)   [64,256]x[256,256]
  gemm_wmma_f32<<<dim3(1, 4), 128, 0, stream>>>(
      p1, FEAT / 2, hw2, FEAT / 2, hb2, p2, FEAT / 2, FEAT / 2, FEAT / 32, 2);
  // pred = p2 @ hw3 + hb3   [64,256]x[256,2]
  pred_head<<<1, 128, 0, stream>>>(p2, hw3, hb3, pred);

  #undef NB
}